// GraphSageMultiClass_48086453846344
// MI455X (gfx1250) — compile-verified
//
#include <hip/hip_runtime.h>
#include <math.h>

typedef __attribute__((ext_vector_type(2))) float v2f;
typedef __attribute__((ext_vector_type(8))) float v8f;

#define N_NODES 100000
#define N_EDGES 1600000
#define F_IN    50
#define HID     128
#define N_GRAPH 64
#define D1_SZ   256
#define D2_SZ   128
#define NCLS    10
#define BN_EPS  1e-3f
#define L2_EPS  1e-12f
#define LDA_PAD 4          // LDS row padding (floats) to avoid bank conflicts

// ---------------------------------------------------------------- utilities
__global__ void fill_f32(float* __restrict__ p, float v, int n) {
    int i = blockIdx.x * blockDim.x + threadIdx.x;
    if (i < n) p[i] = v;
}

__global__ void deg_kernel(const int* __restrict__ row, float* __restrict__ deg, int E) {
    int e = blockIdx.x * blockDim.x + threadIdx.x;
    if (e < E) atomicAdd(&deg[row[e]], 1.0f);
}

// one thread per (edge, feature): agg[row[e], f] += x[col[e], f]
__global__ void scatter_kernel(const float* __restrict__ x,
                               const int* __restrict__ row,
                               const int* __restrict__ col,
                               float* __restrict__ agg, int E, int Fin) {
    int idx = blockIdx.x * blockDim.x + threadIdx.x;   // < E*Fin (<= 204.8M)
    if (idx >= E * Fin) return;
    int e = idx / Fin;
    int f = idx - e * Fin;
    atomicAdd(&agg[(size_t)row[e] * Fin + f], x[(size_t)col[e] * Fin + f]);
}

// ------------------------------------------------- fused SAGE conv + BN layer
// Hout[n, 0..127] = BN(relu(l2norm_row( [Xin | AggSum/max(deg,1)] @ W + b )))
// Block = 256 threads (8 wave32). Block computes 16 rows x 128 cols.
// The 16 x K2 A tile is staged in LDS once per block (mean-normalization of the
// aggregated half fused into the staging load); wave w owns columns
// [16w, 16w+16) and runs the K loop with V_WMMA_F32_16X16X4_F32.
// f32 A fragment layout (ISA 7.12.2): lanes 0-15 hold K=0 (v0) / K=1 (v1),
// lanes 16-31 hold K=2 / K=3, with M = lane&15. B mirrors with N = lane&15.
__global__ void __launch_bounds__(256)
sage_layer_kernel(const float* __restrict__ Xin, const float* __restrict__ AggSum,
                  const float* __restrict__ deg,
                  const float* __restrict__ W,   const float* __restrict__ bias,
                  const float* __restrict__ gam, const float* __restrict__ bet,
                  const float* __restrict__ mu,  const float* __restrict__ var,
                  float* __restrict__ Hout, int Fin) {
    const int K2  = 2 * Fin;                 // 100 / 256 / 256 (all % 4 == 0)
    const int lda = K2 + LDA_PAD;
    const int tid  = threadIdx.x;
    const int wave = tid >> 5;
    const int lane = tid & 31;
    const int m    = lane & 15;              // A row / B-C-D column within tile
    const int hi   = lane >> 4;              // 0 or 1 (K pair / output row half)
    const int row0 = blockIdx.x * 16;
    const int col0 = wave * 16;

    __shared__ float As[16 * (2 * HID + LDA_PAD)];   // 16 x (K2max + pad)
    __shared__ float rowsq[16];

    // ---- stage A tile: [Xin | AggSum/deg] for rows row0..row0+15
    for (int idx = tid; idx < 16 * K2; idx += 256) {
        int r = idx / K2, k = idx - r * K2;
        float v;
        if (k < Fin) v = Xin[(size_t)(row0 + r) * Fin + k];
        else         v = AggSum[(size_t)(row0 + r) * Fin + (k - Fin)]
                         / fmaxf(deg[row0 + r], 1.0f);
        As[r * lda + k] = v;
    }
    if (tid < 16) rowsq[tid] = 0.0f;
    __syncthreads();

    v8f acc = {};
    const float* arow = &As[m * lda];
    for (int k0 = 0; k0 < K2; k0 += 4) {
        int ka = k0 + hi * 2;
        v2f a, b;
        a.x = arow[ka];
        a.y = arow[ka + 1];
        b.x = W[(size_t)ka       * HID + col0 + m];
        b.y = W[(size_t)(ka + 1) * HID + col0 + m];
        acc = __builtin_amdgcn_wmma_f32_16x16x4_f32(
                  false, a, false, b, (short)0, acc, false, false);
    }

    const float bc = bias[col0 + m];
    float vals[8];
#pragma unroll
    for (int j = 0; j < 8; ++j) vals[j] = acc[j] + bc;

    // per-row sum of squares across all 128 columns (8 waves -> LDS atomics)
#pragma unroll
    for (int j = 0; j < 8; ++j) atomicAdd(&rowsq[j + 8 * hi], vals[j] * vals[j]);
    __syncthreads();

    const float bnscale = gam[col0 + m] * rsqrtf(var[col0 + m] + BN_EPS);
    const float bnm = mu[col0 + m], bnb = bet[col0 + m];
#pragma unroll
    for (int j = 0; j < 8; ++j) {
        int r = j + 8 * hi;
        float h = vals[j] * rsqrtf(fmaxf(rowsq[r], L2_EPS));   // l2 normalize
        h = fmaxf(h, 0.0f);                                    // relu
        h = (h - bnm) * bnscale + bnb;                         // batchnorm
        Hout[(size_t)(row0 + r) * HID + col0 + m] = h;
    }
}

// ---------------------------------------------------------------- pooling
__global__ void pool_kernel(const float* __restrict__ H3, const int* __restrict__ seg,
                            float* __restrict__ pool, float* __restrict__ cnt, int Nn) {
    int idx = blockIdx.x * blockDim.x + threadIdx.x;
    int n = idx >> 7, f = idx & 127;
    if (n >= Nn) return;
    int s = seg[n];
    atomicAdd(&pool[s * HID + f], H3[(size_t)n * HID + f]);
    if (f == 0) atomicAdd(&cnt[s], 1.0f);
}

__global__ void pooldiv_kernel(float* __restrict__ pool, const float* __restrict__ cnt) {
    int idx = blockIdx.x * blockDim.x + threadIdx.x;
    if (idx >= N_GRAPH * HID) return;
    pool[idx] /= fmaxf(cnt[idx >> 7], 1.0f);
}

// ----------------------------------------- dense: C = relu(A @ W + b), WMMA f32
// One wave per 16x16 tile. M=64; K=128/256 (multiples of 4).
__global__ void __launch_bounds__(32)
dense_relu_wmma(const float* __restrict__ A, const float* __restrict__ W,
                const float* __restrict__ bias, float* __restrict__ C,
                int K, int Nout) {
    const int lane = threadIdx.x;
    const int m = lane & 15, hi = lane >> 4;
    const int row0 = blockIdx.x * 16, col0 = blockIdx.y * 16;
    v8f acc = {};
    for (int k0 = 0; k0 < K; k0 += 4) {
        int ka = k0 + hi * 2;
        v2f a, b;
        a.x = A[(size_t)(row0 + m) * K + ka];
        a.y = A[(size_t)(row0 + m) * K + ka + 1];
        b.x = W[(size_t)ka       * Nout + col0 + m];
        b.y = W[(size_t)(ka + 1) * Nout + col0 + m];
        acc = __builtin_amdgcn_wmma_f32_16x16x4_f32(
                  false, a, false, b, (short)0, acc, false, false);
    }
    const float bc = bias[col0 + m];
#pragma unroll
    for (int j = 0; j < 8; ++j) {
        int r = j + 8 * hi;
        C[(size_t)(row0 + r) * Nout + col0 + m] = fmaxf(acc[j] + bc, 0.0f);
    }
}

// ------------------------------------------------ head: logits + row softmax
__global__ void __launch_bounds__(32)
head_kernel(const float* __restrict__ Z2, const float* __restrict__ W,
            const float* __restrict__ b, float* __restrict__ out) {
    const int g = blockIdx.x, lane = threadIdx.x;
    float l = 0.0f;
    if (lane < NCLS) {
        l = b[lane];
        for (int k = 0; k < D2_SZ; ++k)
            l += Z2[g * D2_SZ + k] * W[k * NCLS + lane];
    }
    float v = (lane < NCLS) ? l : -INFINITY;
    for (int off = 16; off; off >>= 1) v = fmaxf(v, __shfl_xor(v, off, 32));
    float e = (lane < NCLS) ? expf(l - v) : 0.0f;
    float s = e;
    for (int off = 16; off; off >>= 1) s += __shfl_xor(s, off, 32);
    if (lane < NCLS) out[g * NCLS + lane] = e / s;
}

// ---------------------------------------------------------------- launcher
extern "C" void kernel_launch(void* const* d_in, const int* in_sizes, int n_in,
                              void* d_out, int out_size, void* d_ws, size_t ws_size,
                              hipStream_t stream) {
    const float* x   = (const float*)d_in[0];
    const int* erow  = (const int*)d_in[1];
    const int* ecol  = (const int*)d_in[2];
    const int* seg   = (const int*)d_in[3];
    const float* w1  = (const float*)d_in[4];  const float* b1  = (const float*)d_in[5];
    const float* w2  = (const float*)d_in[6];  const float* b2  = (const float*)d_in[7];
    const float* w3  = (const float*)d_in[8];  const float* b3  = (const float*)d_in[9];
    const float* g1  = (const float*)d_in[10]; const float* be1 = (const float*)d_in[11];
    const float* m1  = (const float*)d_in[12]; const float* v1  = (const float*)d_in[13];
    const float* g2  = (const float*)d_in[14]; const float* be2 = (const float*)d_in[15];
    const float* m2  = (const float*)d_in[16]; const float* v2  = (const float*)d_in[17];
    const float* g3  = (const float*)d_in[18]; const float* be3 = (const float*)d_in[19];
    const float* m3  = (const float*)d_in[20]; const float* v3  = (const float*)d_in[21];
    const float* dw1 = (const float*)d_in[22]; const float* db1 = (const float*)d_in[23];
    const float* dw2 = (const float*)d_in[24]; const float* db2 = (const float*)d_in[25];
    const float* dw3 = (const float*)d_in[26]; const float* db3 = (const float*)d_in[27];
    float* out = (float*)d_out;

    // workspace layout (floats)
    float* ws = (float*)d_ws;
    size_t o = 0;
    float* deg  = ws + o; o += 100352;                    // N, padded
    float* agg  = ws + o; o += (size_t)N_NODES * HID;
    float* hA   = ws + o; o += (size_t)N_NODES * HID;
    float* hB   = ws + o; o += (size_t)N_NODES * HID;
    float* pool = ws + o; o += N_GRAPH * HID;
    float* cnt  = ws + o; o += 256;
    float* z1   = ws + o; o += N_GRAPH * D1_SZ;
    float* z2   = ws + o; o += N_GRAPH * D2_SZ;

    const int TB = 256;
    auto blks = [](long long n, int tb) { return (unsigned)((n + tb - 1) / tb); };

    // in-degree (same for every layer; mean-divide is fused into the GEMM stage)
    fill_f32<<<blks(N_NODES, TB), TB, 0, stream>>>(deg, 0.0f, N_NODES);
    deg_kernel<<<blks(N_EDGES, TB), TB, 0, stream>>>(erow, deg, N_EDGES);

    const unsigned rowTiles = N_NODES / 16;   // 6250

    // ---- layer 1 (Fin = 50)
    fill_f32<<<blks((long long)N_NODES * F_IN, TB), TB, 0, stream>>>(agg, 0.0f, N_NODES * F_IN);
    scatter_kernel<<<blks((long long)N_EDGES * F_IN, TB), TB, 0, stream>>>(x, erow, ecol, agg, N_EDGES, F_IN);
    sage_layer_kernel<<<rowTiles, 256, 0, stream>>>(x, agg, deg, w1, b1, g1, be1, m1, v1, hA, F_IN);

    // ---- layer 2 (Fin = 128)
    fill_f32<<<blks((long long)N_NODES * HID, TB), TB, 0, stream>>>(agg, 0.0f, N_NODES * HID);
    scatter_kernel<<<blks((long long)N_EDGES * HID, TB), TB, 0, stream>>>(hA, erow, ecol, agg, N_EDGES, HID);
    sage_layer_kernel<<<rowTiles, 256, 0, stream>>>(hA, agg, deg, w2, b2, g2, be2, m2, v2, hB, HID);

    // ---- layer 3 (Fin = 128), output reuses hA
    fill_f32<<<blks((long long)N_NODES * HID, TB), TB, 0, stream>>>(agg, 0.0f, N_NODES * HID);
    scatter_kernel<<<blks((long long)N_EDGES * HID, TB), TB, 0, stream>>>(hB, erow, ecol, agg, N_EDGES, HID);
    sage_layer_kernel<<<rowTiles, 256, 0, stream>>>(hB, agg, deg, w3, b3, g3, be3, m3, v3, hA, HID);

    // ---- segment mean pool -> [64, 128]
    fill_f32<<<blks(N_GRAPH * HID + 256, TB), TB, 0, stream>>>(pool, 0.0f, N_GRAPH * HID + 256);
    pool_kernel<<<blks((long long)N_NODES * HID, TB), TB, 0, stream>>>(hA, seg, pool, cnt, N_NODES);
    pooldiv_kernel<<<blks(N_GRAPH * HID, TB), TB, 0, stream>>>(pool, cnt);

    // ---- MLP head
    dense_relu_wmma<<<dim3(N_GRAPH / 16, D1_SZ / 16), 32, 0, stream>>>(pool, dw1, db1, z1, HID,   D1_SZ);
    dense_relu_wmma<<<dim3(N_GRAPH / 16, D2_SZ / 16), 32, 0, stream>>>(z1,   dw2, db2, z2, D1_SZ, D2_SZ);
    head_kernel<<<N_GRAPH, 32, 0, stream>>>(z2, dw3, db3, out);
}